// iterativeGCN_inductive_64278480552431
// MI455X (gfx1250) — compile-verified
//
#include <hip/hip_runtime.h>
#include <hip/hip_bf16.h>

#define N_NODES  150000
#define N_EDGES  300000
#define HIDDEN   256
#define N_GRAPHS 6000
#define NUM_TASKS 128
#define N_FEATS  9
#define VOCAB    128
#define BN_EPS   1e-5f
#define NPAD     150016   // 128-row multiple for GEMM tiling

typedef __attribute__((ext_vector_type(2))) float v2f;
typedef __attribute__((ext_vector_type(8))) float v8f;

// ---------------- AtomEncoder: x[n,d] = sum_f emb[f, xa[n,f], d] ----------------
__global__ __launch_bounds__(HIDDEN) void k_encode(const int* __restrict__ xa,
                                                   const float* __restrict__ emb,
                                                   float* __restrict__ x) {
  const int n = blockIdx.x;
  const int d = threadIdx.x;
  float acc = 0.f;
  if (n < N_NODES) {
#pragma unroll
    for (int f = 0; f < N_FEATS; ++f) {
      const int idx = xa[n * N_FEATS + f];
      acc += emb[(f * VOCAB + idx) * HIDDEN + d];
    }
  }
  x[(size_t)n * HIDDEN + d] = acc;   // pad rows -> 0
}

// ---------------- small utilities ----------------
__global__ void k_zero_i32(int* __restrict__ p, int n) {
  const int i = blockIdx.x * blockDim.x + threadIdx.x;
  if (i < n) p[i] = 0;
}

__global__ void k_count(const int* __restrict__ ei, int* __restrict__ indeg) {
  const int e = blockIdx.x * blockDim.x + threadIdx.x;
  if (e < N_EDGES) atomicAdd(&indeg[ei[N_EDGES + e]], 1);
}

__global__ void k_dinv(const int* __restrict__ indeg, float* __restrict__ dinv) {
  const int n = blockIdx.x * blockDim.x + threadIdx.x;
  if (n < N_NODES) dinv[n] = rsqrtf((float)indeg[n] + 1.0f);  // +1 = self loop
}

// transpose W (256x256, k-major) -> Wt (n-major) so B-fragments are contiguous b64
__global__ __launch_bounds__(256) void k_transpose(const float* __restrict__ W,
                                                   float* __restrict__ Wt) {
  const int i = blockIdx.x * blockDim.x + threadIdx.x;   // 65536 threads
  const int n = i & (HIDDEN - 1);
  const int k = i >> 8;
  Wt[(size_t)n * HIDDEN + k] = W[(size_t)k * HIDDEN + n];
}

// ---------------- exclusive scan of indeg -> csr row pointers ----------------
#define SCAN_B 256
__global__ __launch_bounds__(SCAN_B) void k_scan1(const int* __restrict__ in,
                                                  int* __restrict__ outInc,
                                                  int* __restrict__ bsum, int n) {
  __shared__ int sh[SCAN_B];
  const int gid = blockIdx.x * SCAN_B + threadIdx.x;
  sh[threadIdx.x] = (gid < n) ? in[gid] : 0;
  __syncthreads();
  for (int off = 1; off < SCAN_B; off <<= 1) {
    const int t = (threadIdx.x >= off) ? sh[threadIdx.x - off] : 0;
    __syncthreads();
    sh[threadIdx.x] += t;
    __syncthreads();
  }
  if (gid < n) outInc[gid] = sh[threadIdx.x];
  if (threadIdx.x == SCAN_B - 1) bsum[blockIdx.x] = sh[threadIdx.x];
}

__global__ void k_scan2(int* __restrict__ bsum, int nb, int* __restrict__ ptrEnd) {
  if (blockIdx.x == 0 && threadIdx.x == 0) {
    int run = 0;
    for (int b = 0; b < nb; ++b) { const int t = bsum[b]; bsum[b] = run; run += t; }
    *ptrEnd = run;   // == N_EDGES
  }
}

__global__ __launch_bounds__(SCAN_B) void k_scan3(const int* __restrict__ inc,
                                                  const int* __restrict__ indeg,
                                                  const int* __restrict__ bsum,
                                                  int* __restrict__ ptr, int n) {
  const int i = blockIdx.x * SCAN_B + threadIdx.x;
  if (i < n) ptr[i] = inc[i] - indeg[i] + bsum[blockIdx.x];  // exclusive
}

__global__ void k_fill(const int* __restrict__ ei, const int* __restrict__ ptr,
                       int* __restrict__ fillcnt, const float* __restrict__ dinv,
                       int* __restrict__ csrc, float* __restrict__ cw) {
  const int e = blockIdx.x * blockDim.x + threadIdx.x;
  if (e >= N_EDGES) return;
  const int s = ei[e];
  const int c = ei[N_EDGES + e];
  const int pos = ptr[c] + atomicAdd(&fillcnt[c], 1);
  csrc[pos] = s;
  cw[pos] = dinv[s] * dinv[c];
}

// ---------------- fp32 WMMA GEMM: M = X @ W  (NPAD x 256) @ (256 x 256) ----------------
// Wt is W transposed (n-major): lane's B-fragment = Wt[n][k],Wt[n][k+1] -> one b64 load.
#define GEMM_ROWS 128
#define LDA 260   // pad: A-frag ds_load_b64 hits 32 distinct bank pairs
__global__ __launch_bounds__(256) void k_gemm(const float* __restrict__ X,
                                              const float* __restrict__ Wt,
                                              float* __restrict__ Mout) {
  __shared__ float As[GEMM_ROWS * LDA];   // 130 KB of the 320 KB WGP LDS
  const int r0 = blockIdx.x * GEMM_ROWS;
  const int tid = threadIdx.x;

  // cooperative stage of the 128x256 A tile (float4, fully coalesced)
  for (int i = tid; i < GEMM_ROWS * (HIDDEN / 4); i += 256) {
    const int row = i >> 6;          // 64 float4 per row
    const int c4 = (i & 63) << 2;
    const float4 v = *(const float4*)(X + (size_t)(r0 + row) * HIDDEN + c4);
    float* dst = &As[row * LDA + c4];
    dst[0] = v.x; dst[1] = v.y; dst[2] = v.z; dst[3] = v.w;
  }
  __syncthreads();

  const int wave = tid >> 5;
  const int lane = tid & 31;
  const int mrow = wave * 16 + (lane & 15);   // A: M = lane%16
  const int khalf = (lane >> 4) << 1;         // A/B: lanes 16-31 hold K+2,K+3
  const int ncol = lane & 15;                 // B/C/D: N = lane%16

  for (int ng = 0; ng < 4; ++ng) {
    v8f acc0 = {}; v8f acc1 = {}; v8f acc2 = {}; v8f acc3 = {};
    const int nb = ng * 64;
    const float* w0 = Wt + (size_t)(nb + ncol) * HIDDEN + khalf;
    for (int k0 = 0; k0 < HIDDEN; k0 += 4) {
      v2f a;
      a.x = As[mrow * LDA + k0 + khalf];
      a.y = As[mrow * LDA + k0 + khalf + 1];
      const v2f b0 = *(const v2f*)(w0 + k0);
      const v2f b1 = *(const v2f*)(w0 + 16 * HIDDEN + k0);
      const v2f b2 = *(const v2f*)(w0 + 32 * HIDDEN + k0);
      const v2f b3 = *(const v2f*)(w0 + 48 * HIDDEN + k0);
      acc0 = __builtin_amdgcn_wmma_f32_16x16x4_f32(false, a, false, b0, (short)0, acc0, false, false);
      acc1 = __builtin_amdgcn_wmma_f32_16x16x4_f32(false, a, false, b1, (short)0, acc1, false, false);
      acc2 = __builtin_amdgcn_wmma_f32_16x16x4_f32(false, a, false, b2, (short)0, acc2, false, false);
      acc3 = __builtin_amdgcn_wmma_f32_16x16x4_f32(false, a, false, b3, (short)0, acc3, false, false);
    }
    // D layout: VGPR v -> M = v + 8*(lane>=16), N = lane%16
    const int mbase = r0 + wave * 16 + ((lane >> 4) << 3);
#pragma unroll
    for (int v = 0; v < 8; ++v) {
      Mout[(size_t)(mbase + v) * HIDDEN + nb + ncol]      = acc0[v];
      Mout[(size_t)(mbase + v) * HIDDEN + nb + 16 + ncol] = acc1[v];
      Mout[(size_t)(mbase + v) * HIDDEN + nb + 32 + ncol] = acc2[v];
      Mout[(size_t)(mbase + v) * HIDDEN + nb + 48 + ncol] = acc3[v];
    }
  }
}

// ---------------- fused: aggregate + bias + ReLU + folded-BN + residual ----------------
__global__ __launch_bounds__(HIDDEN) void k_update(const float* __restrict__ Mm,
    const int* __restrict__ ptr, const int* __restrict__ csrc, const float* __restrict__ cw,
    const float* __restrict__ dinv, const float* __restrict__ b_conv,
    const float* __restrict__ gamma, const float* __restrict__ beta,
    const float* __restrict__ mean, const float* __restrict__ var,
    float* __restrict__ x, float sc) {
  const int n = blockIdx.x;
  const int d = threadIdx.x;
  const float di = dinv[n];
  float agg = b_conv[d] + di * di * Mm[(size_t)n * HIDDEN + d];   // self loop
  const int p0 = ptr[n], p1 = ptr[n + 1];
  for (int i = p0; i < p1; ++i) {
    const int s = csrc[i];
    agg += cw[i] * Mm[(size_t)s * HIDDEN + d];
  }
  const float istd = gamma[d] * rsqrtf(var[d] + BN_EPS);
  const float sh = beta[d] - mean[d] * istd;
  const float bn = fmaxf(agg, 0.f) * istd + sh;
  const size_t xi = (size_t)n * HIDDEN + d;
  x[xi] = sc * x[xi] + (1.f - sc) * bn;
}

// ---------------- pooling ----------------
__global__ void k_bounds(const int* __restrict__ batch, int* __restrict__ gstart) {
  const int g = blockIdx.x * blockDim.x + threadIdx.x;
  if (g > N_GRAPHS) return;
  int lo = 0, hi = N_NODES;
  while (lo < hi) { const int mid = (lo + hi) >> 1; if (batch[mid] < g) lo = mid + 1; else hi = mid; }
  gstart[g] = lo;
}

__global__ __launch_bounds__(HIDDEN) void k_pool(const float* __restrict__ x,
                                                 const int* __restrict__ gstart,
                                                 float* __restrict__ pooled) {
  const int g = blockIdx.x;
  const int d = threadIdx.x;
  const int n0 = gstart[g], n1 = gstart[g + 1];
  float s = 0.f;
  for (int n = n0; n < n1; ++n) s += x[(size_t)n * HIDDEN + d];
  pooled[g * HIDDEN + d] = s / fmaxf((float)(n1 - n0), 1.f);
}

__global__ __launch_bounds__(NUM_TASKS) void k_pred(const float* __restrict__ pooled,
                                                    const float* __restrict__ Wp,
                                                    const float* __restrict__ bp,
                                                    float* __restrict__ out) {
  const int g = blockIdx.x;
  const int t = threadIdx.x;
  float acc = bp[t];
  for (int k = 0; k < HIDDEN; ++k)
    acc += pooled[g * HIDDEN + k] * Wp[k * NUM_TASKS + t];
  out[g * NUM_TASKS + t] = acc;
}

// ---------------- host ----------------
extern "C" void kernel_launch(void* const* d_in, const int* in_sizes, int n_in,
                              void* d_out, int out_size, void* d_ws, size_t ws_size,
                              hipStream_t stream) {
  const int*   xa      = (const int*)d_in[0];
  const int*   ei      = (const int*)d_in[1];
  const int*   batch   = (const int*)d_in[2];
  const float* emb     = (const float*)d_in[3];
  const float* W       = (const float*)d_in[4];
  const float* b_conv  = (const float*)d_in[5];
  const float* gamma   = (const float*)d_in[6];
  const float* beta    = (const float*)d_in[7];
  const float* bn_mean = (const float*)d_in[8];
  const float* bn_var  = (const float*)d_in[9];
  const float* W_pred  = (const float*)d_in[10];
  const float* b_pred  = (const float*)d_in[11];
  float* out = (float*)d_out;

  char* ws = (char*)d_ws;
  size_t off = 0;
  auto alloc = [&](size_t bytes) { char* p = ws + off; off = (off + bytes + 255) & ~(size_t)255; return p; };
  float* X      = (float*)alloc((size_t)NPAD * HIDDEN * 4);
  float* Mm     = (float*)alloc((size_t)NPAD * HIDDEN * 4);
  float* Wt     = (float*)alloc((size_t)HIDDEN * HIDDEN * 4);
  float* dinv   = (float*)alloc((size_t)N_NODES * 4);
  int*   indeg  = (int*)  alloc((size_t)N_NODES * 4);
  int*   ptr    = (int*)  alloc((size_t)(N_NODES + 1) * 4);
  int*   scanI  = (int*)  alloc((size_t)N_NODES * 4);
  int*   bsum   = (int*)  alloc(1024 * 4);
  int*   fillc  = (int*)  alloc((size_t)N_NODES * 4);
  int*   csrc   = (int*)  alloc((size_t)N_EDGES * 4);
  float* cw     = (float*)alloc((size_t)N_EDGES * 4);
  int*   gstart = (int*)  alloc((size_t)(N_GRAPHS + 1) * 4);
  float* pooled = (float*)alloc((size_t)N_GRAPHS * HIDDEN * 4);
  (void)in_sizes; (void)n_in; (void)out_size; (void)ws_size;

  const int nbN = (N_NODES + SCAN_B - 1) / SCAN_B;
  const int nbE = (N_EDGES + 255) / 256;

  // encode + W transpose + degree/CSR build (once)
  k_encode<<<NPAD, HIDDEN, 0, stream>>>(xa, emb, X);
  k_transpose<<<(HIDDEN * HIDDEN) / 256, 256, 0, stream>>>(W, Wt);
  k_zero_i32<<<nbN, 256, 0, stream>>>(indeg, N_NODES);
  k_zero_i32<<<nbN, 256, 0, stream>>>(fillc, N_NODES);
  k_count<<<nbE, 256, 0, stream>>>(ei, indeg);
  k_dinv<<<nbN, 256, 0, stream>>>(indeg, dinv);
  k_scan1<<<nbN, SCAN_B, 0, stream>>>(indeg, scanI, bsum, N_NODES);
  k_scan2<<<1, 1, 0, stream>>>(bsum, nbN, ptr + N_NODES);
  k_scan3<<<nbN, SCAN_B, 0, stream>>>(scanI, indeg, bsum, ptr, N_NODES);
  k_fill<<<nbE, 256, 0, stream>>>(ei, ptr, fillc, dinv, csrc, cw);

  // 8 smoothed GCN iterations
  const float sched[8] = {0.9f, 0.85f, 0.8f, 0.75f, 0.7f, 0.65f, 0.6f, 0.55f};
  for (int it = 0; it < 8; ++it) {
    k_gemm<<<NPAD / GEMM_ROWS, 256, 0, stream>>>(X, Wt, Mm);
    k_update<<<N_NODES, HIDDEN, 0, stream>>>(Mm, ptr, csrc, cw, dinv, b_conv,
                                             gamma, beta, bn_mean, bn_var, X, sched[it]);
  }

  // mean pool + prediction head
  k_bounds<<<(N_GRAPHS + 256) / 256, 256, 0, stream>>>(batch, gstart);
  k_pool<<<N_GRAPHS, HIDDEN, 0, stream>>>(X, gstart, pooled);
  k_pred<<<N_GRAPHS, NUM_TASKS, 0, stream>>>(pooled, W_pred, b_pred, out);
}